// GINStateEncoder_61555471286657
// MI455X (gfx1250) — compile-verified
//
#include <hip/hip_runtime.h>

#define N_NODES_  100000
#define N_EDGES_  600000
#define N_GRAPHS_ 64

typedef __attribute__((ext_vector_type(2))) float v2f;
typedef __attribute__((ext_vector_type(8))) float v8f;

// ---------------------------------------------------------------------------
// Vectorized copy (h_pre = x)
// ---------------------------------------------------------------------------
__global__ void copy_f4(float4* __restrict__ d, const float4* __restrict__ s, int n4) {
    int i = blockIdx.x * blockDim.x + threadIdx.x;
    if (i < n4) d[i] = s[i];
}

// ---------------------------------------------------------------------------
// Edge scatter-add: acc[dst[e]] += feat[src[e]]   (one thread per edge x 4ch)
// ---------------------------------------------------------------------------
template<int D>
__global__ void edge_aggr_kernel(float* __restrict__ acc, const float* __restrict__ feat,
                                 const long long* __restrict__ src,
                                 const long long* __restrict__ dst, int n_edges) {
    constexpr int CH = D / 4;
    int i = blockIdx.x * blockDim.x + threadIdx.x;
    int total = n_edges * CH;
    if (i >= total) return;
    int e = i / CH;
    int c = (i % CH) * 4;
    long long s = src[e];
    long long d = dst[e];
    const float4 v = *(const float4*)(feat + (size_t)s * D + c);
    float* o = acc + (size_t)d * D + c;
    atomicAdd(o + 0, v.x);
    atomicAdd(o + 1, v.y);
    atomicAdd(o + 2, v.z);
    atomicAdd(o + 3, v.w);
}

// ---------------------------------------------------------------------------
// Fused GIN MLP: hout = relu(bn(relu(hpre @ w1 + b1) @ w2 + b2))
// 64 rows per block, 8 waves, fp32 WMMA 16x16x4.
// ---------------------------------------------------------------------------
template<int D_IN, int D_OUT>
__global__ __launch_bounds__(256) void gin_mlp_kernel(
    const float* __restrict__ hpre,   // N x D_IN
    const float* __restrict__ w1,     // D_IN x 64
    const float* __restrict__ b1,     // 64
    const float* __restrict__ w2,     // 64 x D_OUT
    const float* __restrict__ b2,     // D_OUT
    const float* __restrict__ bng, const float* __restrict__ bnb,
    const float* __restrict__ bnm, const float* __restrict__ bnv,
    float* __restrict__ hout,         // N x D_OUT
    int n_nodes)
{
    constexpr int ROWS = 64;
    constexpr int DH   = 64;
    __shared__ float sA[ROWS * D_IN];   // activation tile
    __shared__ float sH[ROWS * DH];     // hidden tile

    const int tid    = threadIdx.x;
    const int wave   = tid >> 5;
    const int lane   = tid & 31;
    const int lane16 = lane & 15;
    const int hi     = lane >> 4;       // 0: K=0,1 / rows 0-7 ; 1: K=2,3 / rows 8-15
    const int row0   = blockIdx.x * ROWS;

    // Stage activation tile (zero-fill OOB rows so WMMA math is safe)
    for (int i = tid; i < ROWS * D_IN; i += 256) {
        int r = i / D_IN;
        int c = i % D_IN;
        sA[i] = (row0 + r < n_nodes) ? hpre[(size_t)(row0 + r) * D_IN + c] : 0.0f;
    }
    __syncthreads();

    // ---- GEMM1: (ROWS x D_IN) @ (D_IN x 64) + b1, ReLU -> sH ----
    constexpr int MT  = ROWS / 16;   // 4
    constexpr int NT1 = DH / 16;     // 4
    for (int t = wave; t < MT * NT1; t += 8) {
        int mt = t / NT1, nt = t % NT1;
        v8f acc = {};
        const float* ap = &sA[(mt * 16 + lane16) * D_IN + hi * 2];
        const float* bp = &w1[(hi * 2) * DH + nt * 16 + lane16];
        #pragma unroll
        for (int k = 0; k < D_IN; k += 4) {
            v2f a = *(const v2f*)(ap + k);
            v2f b;
            b.x = bp[k * DH];
            b.y = bp[k * DH + DH];
            acc = __builtin_amdgcn_wmma_f32_16x16x4_f32(false, a, false, b,
                                                        (short)0, acc, false, false);
        }
        int col = nt * 16 + lane16;
        float bias = b1[col];
        #pragma unroll
        for (int r = 0; r < 8; r++) {
            float v = acc[r] + bias;
            sH[(mt * 16 + hi * 8 + r) * DH + col] = v > 0.0f ? v : 0.0f;
        }
    }
    __syncthreads();

    // ---- GEMM2: (ROWS x 64) @ (64 x D_OUT) + b2, BN, ReLU -> global ----
    constexpr int NT2 = D_OUT / 16;
    for (int t = wave; t < MT * NT2; t += 8) {
        int mt = t / NT2, nt = t % NT2;
        v8f acc = {};
        const float* ap = &sH[(mt * 16 + lane16) * DH + hi * 2];
        const float* bp = &w2[(hi * 2) * D_OUT + nt * 16 + lane16];
        #pragma unroll
        for (int k = 0; k < DH; k += 4) {
            v2f a = *(const v2f*)(ap + k);
            v2f b;
            b.x = bp[k * D_OUT];
            b.y = bp[k * D_OUT + D_OUT];
            acc = __builtin_amdgcn_wmma_f32_16x16x4_f32(false, a, false, b,
                                                        (short)0, acc, false, false);
        }
        int col = nt * 16 + lane16;
        float scale = bng[col] * rsqrtf(bnv[col] + 1e-5f);
        float shift = (b2[col] - bnm[col]) * scale + bnb[col];
        #pragma unroll
        for (int r = 0; r < 8; r++) {
            int row = row0 + mt * 16 + hi * 8 + r;
            if (row < n_nodes) {
                float v = acc[r] * scale + shift;
                hout[(size_t)row * D_OUT + col] = v > 0.0f ? v : 0.0f;
            }
        }
    }
}

// ---------------------------------------------------------------------------
// Global mean pool over graphs
// ---------------------------------------------------------------------------
__global__ void pool_zero(float* __restrict__ out, float* __restrict__ cnt) {
    int i = blockIdx.x * blockDim.x + threadIdx.x;
    if (i < N_GRAPHS_ * 96) out[i] = 0.0f;
    if (i < N_GRAPHS_) cnt[i] = 0.0f;
}

__global__ void pool_scatter(const float* __restrict__ h, const long long* __restrict__ batch,
                             float* __restrict__ out, float* __restrict__ cnt) {
    int i = blockIdx.x * blockDim.x + threadIdx.x;
    const int total = N_NODES_ * 24;           // 96/4 float4 chunks per node
    if (i >= total) return;
    int node = i / 24;
    int c4   = i % 24;
    long long g = batch[node];
    const float4 v = *(const float4*)(h + (size_t)node * 96 + c4 * 4);
    float* o = out + (size_t)g * 96 + c4 * 4;
    atomicAdd(o + 0, v.x);
    atomicAdd(o + 1, v.y);
    atomicAdd(o + 2, v.z);
    atomicAdd(o + 3, v.w);
    if (c4 == 0) atomicAdd(cnt + g, 1.0f);
}

__global__ void pool_div(float* __restrict__ out, const float* __restrict__ cnt) {
    int i = blockIdx.x * blockDim.x + threadIdx.x;
    if (i < N_GRAPHS_ * 96) {
        float c = cnt[i / 96];
        out[i] = out[i] / (c > 1.0f ? c : 1.0f);
    }
}

// ---------------------------------------------------------------------------
// Launch
// ---------------------------------------------------------------------------
extern "C" void kernel_launch(void* const* d_in, const int* in_sizes, int n_in,
                              void* d_out, int out_size, void* d_ws, size_t ws_size,
                              hipStream_t stream) {
    (void)in_sizes; (void)n_in; (void)out_size; (void)ws_size;

    const float*     x     = (const float*)d_in[0];
    const long long* ei    = (const long long*)d_in[1];
    const long long* batch = (const long long*)d_in[2];
    const long long* src   = ei;
    const long long* dst   = ei + N_EDGES_;

    // per layer: w1, b1, w2, b2, bn_g, bn_b, bn_m, bn_v
    const float* L[3][8];
    for (int l = 0; l < 3; ++l)
        for (int j = 0; j < 8; ++j)
            L[l][j] = (const float*)d_in[3 + l * 8 + j];

    float* A   = (float*)d_ws;                       // 100000 x 128
    float* B   = A + (size_t)N_NODES_ * 128;         // 100000 x 64
    float* C   = B + (size_t)N_NODES_ * 64;          // 100000 x 96
    float* cnt = C + (size_t)N_NODES_ * 96;          // 64
    float* out = (float*)d_out;                      // 64 x 96

    auto cdiv = [](int a, int b) { return (a + b - 1) / b; };
    const int mlp_blocks = cdiv(N_NODES_, 64);

    // ---- Layer 0 : x(128) -> B(64) ----
    copy_f4<<<cdiv(N_NODES_ * 128 / 4, 256), 256, 0, stream>>>(
        (float4*)A, (const float4*)x, N_NODES_ * 128 / 4);
    edge_aggr_kernel<128><<<cdiv(N_EDGES_ * 32, 256), 256, 0, stream>>>(
        A, x, src, dst, N_EDGES_);
    gin_mlp_kernel<128, 64><<<mlp_blocks, 256, 0, stream>>>(
        A, L[0][0], L[0][1], L[0][2], L[0][3], L[0][4], L[0][5], L[0][6], L[0][7],
        B, N_NODES_);

    // ---- Layer 1 : B(64) -> A(64) ----
    copy_f4<<<cdiv(N_NODES_ * 64 / 4, 256), 256, 0, stream>>>(
        (float4*)C, (const float4*)B, N_NODES_ * 64 / 4);
    edge_aggr_kernel<64><<<cdiv(N_EDGES_ * 16, 256), 256, 0, stream>>>(
        C, B, src, dst, N_EDGES_);
    gin_mlp_kernel<64, 64><<<mlp_blocks, 256, 0, stream>>>(
        C, L[1][0], L[1][1], L[1][2], L[1][3], L[1][4], L[1][5], L[1][6], L[1][7],
        A, N_NODES_);

    // ---- Layer 2 : A(64) -> C(96) ----
    copy_f4<<<cdiv(N_NODES_ * 64 / 4, 256), 256, 0, stream>>>(
        (float4*)B, (const float4*)A, N_NODES_ * 64 / 4);
    edge_aggr_kernel<64><<<cdiv(N_EDGES_ * 16, 256), 256, 0, stream>>>(
        B, A, src, dst, N_EDGES_);
    gin_mlp_kernel<64, 96><<<mlp_blocks, 256, 0, stream>>>(
        B, L[2][0], L[2][1], L[2][2], L[2][3], L[2][4], L[2][5], L[2][6], L[2][7],
        C, N_NODES_);

    // ---- Global mean pool over 64 graphs ----
    pool_zero<<<cdiv(N_GRAPHS_ * 96, 256), 256, 0, stream>>>(out, cnt);
    pool_scatter<<<cdiv(N_NODES_ * 24, 256), 256, 0, stream>>>(C, batch, out, cnt);
    pool_div<<<cdiv(N_GRAPHS_ * 96, 256), 256, 0, stream>>>(out, cnt);
}